// ParamKDE_88055419503227
// MI455X (gfx1250) — compile-verified
//
#include <hip/hip_runtime.h>
#include <hip/hip_bf16.h>
#include <math.h>

// ParamKDE on MI455X (gfx1250):
//   out[m] = log(EPS + exp(t1)/N * sum_n exp(-0.5*||xe[m]-xb[n]||^2 / sigma^2))
// Cross term via full-precision V_WMMA_F32_16X16X4_F32 chains (wave32 tensor op),
// B tiles double-buffered in registers so each tile's 16-load clause overlaps the
// previous tile's WMMA+exp chain. Raw v_exp_f32 (underflow->0 is exact for a sum
// of positive exponentials). LDS ds_add_f32 cross-wave reduction.

typedef float v2f __attribute__((ext_vector_type(2)));
typedef float v8f __attribute__((ext_vector_type(8)));

#define KDE_M 4096
#define KDE_N 8192
#define KDE_D 64

__device__ __forceinline__ void load_tile(v2f (&b)[16], const float* __restrict__ p) {
#pragma unroll
    for (int k = 0; k < 16; ++k)
        b[k] = *(const v2f*)(p + 4 * k);   // global_load_b64, clause-able
}

__device__ __forceinline__ void accum_tile(const v2f (&a)[16], const v2f (&b)[16],
                                           const float (&e2kv)[8], float k2, float m2k,
                                           float (&acc)[8]) {
    // column norm: each lane holds half its column's dims; xor-16 completes it
    float b2 = 0.0f;
#pragma unroll
    for (int k = 0; k < 16; ++k)
        b2 += b[k].x * b[k].x + b[k].y * b[k].y;
    const float xb2k = k2 * (b2 + __shfl_xor(b2, 16, 32));

    // cross[m][n] via 16 chained f32 WMMAs over K=64
    v8f c = {0.f, 0.f, 0.f, 0.f, 0.f, 0.f, 0.f, 0.f};
#pragma unroll
    for (int k = 0; k < 16; ++k) {
        c = __builtin_amdgcn_wmma_f32_16x16x4_f32(
                /*neg_a=*/false, a[k], /*neg_b=*/false, b[k],
                /*c_mod=*/(short)0, c, /*reuse_a=*/false, /*reuse_b=*/false);
    }

    // arg = k2*(xe2 + xb2 - 2*cross) = (k2*xe2 + k2*xb2) + (-2k2)*cross
    // one v_fma_f32 + one bare v_exp_f32 + one add per element
#pragma unroll
    for (int v = 0; v < 8; ++v) {
        const float arg = __builtin_fmaf(m2k, c[v], e2kv[v] + xb2k);
        acc[v] += __builtin_amdgcn_exp2f(arg);
    }
}

__global__ __launch_bounds__(256) void ParamKDE_88055419503227_kernel(
    const float* __restrict__ x_eval,    // [M, 64]
    const float* __restrict__ x_base,    // [N, 64]
    const float* __restrict__ log_sigma, // [1]
    float* __restrict__ out)             // [M]
{
    __shared__ float srow[16];

    const int tid  = threadIdx.x;
    const int wave = tid >> 5;        // 8 waves per workgroup
    const int lane = tid & 31;        // wave32
    const int hi   = lane >> 4;       // lane half selects K-pair (A/B f32 layout)
    const int lm   = lane & 15;       // row (A) / col (B) within tile
    const int rowBase = blockIdx.x * 16;

    if (tid < 16) srow[tid] = 0.0f;
    __syncthreads();

    // Scalar prologue (uniform): sigma^2, exp coefficient, outer scale.
    const float ls     = log_sigma[0];
    const float sigma2 = __expf(2.0f * ls);
    // exp(-0.5*sq/sigma2) == exp2(k2*sq), k2 = -0.5/(sigma2*ln2)
    const float k2     = -0.72134752044448169f / sigma2;
    const float m2k    = -2.0f * k2;
    // t1 = -0.5*D*log(2*pi) - log_sigma ; scale = exp(t1)/N
    const float t1     = -0.5f * (float)KDE_D * 1.8378770664093453f - ls;
    const float scale  = __expf(t1) * (1.0f / (float)KDE_N);

    // ---- Preload A tile: 16 rows x 64 dims, 16x4-f32 WMMA A layout ----
    // lane L (row = L&15): VGPR pair k holds dims {4k+2*hi, 4k+2*hi+1}
    const float* pA = x_eval + (size_t)(rowBase + lm) * KDE_D + 2 * hi;
    v2f a[16];
    float a2 = 0.0f;
#pragma unroll
    for (int k = 0; k < 16; ++k) {
        a[k] = *(const v2f*)(pA + 4 * k);
        a2 += a[k].x * a[k].x + a[k].y * a[k].y;
    }
    const float xe2 = a2 + __shfl_xor(a2, 16, 32);
    // broadcast row norms to C-layout row index (VGPR v -> row v + 8*hi),
    // pre-scaled by k2 so the epilogue is a single fma per element
    float e2kv[8];
#pragma unroll
    for (int v = 0; v < 8; ++v)
        e2kv[v] = k2 * __shfl(xe2, v + 8 * hi, 32);

    float acc[8] = {0.f, 0.f, 0.f, 0.f, 0.f, 0.f, 0.f, 0.f};

    // ---- Each wave sweeps a 1024-column slice of N: 64 tiles of 16 ----
    // Double-buffered B tiles: next tile's load clause issues before the
    // current tile's WMMA+exp chain, hiding L2 latency inside one wave.
    const int colStart = wave * (KDE_N / 8);
    const float* pB0 = x_base + (size_t)(colStart + lm) * KDE_D + 2 * hi;
    const size_t tileStride = (size_t)16 * KDE_D;   // 16 columns of 64 dims

    v2f b0[16], b1[16];
    load_tile(b0, pB0);
    for (int t = 0; t < 64; t += 2) {
        load_tile(b1, pB0 + (size_t)(t + 1) * tileStride);
        accum_tile(a, b0, e2kv, k2, m2k, acc);
        if (t + 2 < 64)
            load_tile(b0, pB0 + (size_t)(t + 2) * tileStride);
        accum_tile(a, b1, e2kv, k2, m2k, acc);
    }

    // ---- Reduce over the 16 columns held across lanes of each half ----
#pragma unroll
    for (int v = 0; v < 8; ++v) {
        float r = acc[v];
        r += __shfl_xor(r, 1, 32);
        r += __shfl_xor(r, 2, 32);
        r += __shfl_xor(r, 4, 32);
        r += __shfl_xor(r, 8, 32);
        if (lm == 0) {
            // row index for VGPR v in this lane-half is v + 8*hi
            atomicAdd(&srow[v + 8 * hi], r);   // ds_add_f32 across 8 waves
        }
    }
    __syncthreads();

    if (tid < 16) {
        out[rowBase + tid] = logf(1e-8f + scale * srow[tid]);
    }
}

extern "C" void kernel_launch(void* const* d_in, const int* in_sizes, int n_in,
                              void* d_out, int out_size, void* d_ws, size_t ws_size,
                              hipStream_t stream) {
    const float* x_eval    = (const float*)d_in[0];  // [4096*64]
    const float* x_base    = (const float*)d_in[1];  // [8192*64]
    const float* log_sigma = (const float*)d_in[2];  // [1]
    float* out             = (float*)d_out;          // [4096]
    (void)in_sizes; (void)n_in; (void)out_size; (void)d_ws; (void)ws_size;

    dim3 grid(KDE_M / 16);   // 256 workgroups, one 16-row tile each
    dim3 block(256);         // 8 waves; each wave owns 1024 columns of N
    ParamKDE_88055419503227_kernel<<<grid, block, 0, stream>>>(
        x_eval, x_base, log_sigma, out);
}